// PeriodicPrimitives2D_39161511805435
// MI455X (gfx1250) — compile-verified
//
#include <hip/hip_runtime.h>

typedef float v2f __attribute__((ext_vector_type(2)));
typedef float v8f __attribute__((ext_vector_type(8)));

#define NWAVES 128
#define NPAIRS (NWAVES / 2)
#define WAVES_PER_BLOCK 8
#define BLOCK (WAVES_PER_BLOCK * 32)

// out[p] = sum_w c0_w*sin(2*pi*f_w*(x0*cos r_w + x1*sin r_w)) + c1_w*cos(...)
//  - per-block LDS tables laid out fragment-shaped per 2-wave pair:
//      ab[pp] = {A(w0), B(w0), A(w1), B(w1)}   A=f*cos(r), B=f*sin(r)
//      cc[pp] = {c0(w0), c0(w1), c1(w0), c1(w1)}  (B-fragments contiguous)
//  - t = x0*A + x1*B ; r = t - rint(t) in [-0.5,0.5] revolutions
//  - hardware sin/cos take revolutions: __builtin_amdgcn_sinf(r) == sin(2*pi*t)
//  - accumulate over 128 waves with V_WMMA_F32_16X16X4_F32 (lossless f32):
//    A-frag = sin/cos values for the 16-point tile (K = vgpr + 2*lane_half),
//    B-frag = coefficient pair replicated across all 16 columns.
__global__ __launch_bounds__(BLOCK) void periodic2d_wmma_kernel(
    const float* __restrict__ x, const float* __restrict__ freqs,
    const float* __restrict__ rots, const float* __restrict__ coeffs,
    float* __restrict__ out, int npts, int ntiles)
{
    __shared__ float4 ab[NPAIRS];
    __shared__ float4 cc[NPAIRS];
    const float inv2pi = 0.15915493667125702f;

    int tid = threadIdx.x;
    if (tid < NPAIRS) {
        int w0 = 2 * tid, w1 = 2 * tid + 1;
        float f0 = freqs[w0], f1 = freqs[w1];
        float r0 = rots[w0] * inv2pi;             // revolutions in [0, 0.5]
        float r1 = rots[w1] * inv2pi;
        ab[tid] = make_float4(f0 * __builtin_amdgcn_cosf(r0),
                              f0 * __builtin_amdgcn_sinf(r0),
                              f1 * __builtin_amdgcn_cosf(r1),
                              f1 * __builtin_amdgcn_sinf(r1));
        cc[tid] = make_float4(coeffs[2 * w0 + 0], coeffs[2 * w1 + 0],
                              coeffs[2 * w0 + 1], coeffs[2 * w1 + 1]);
    }
    __syncthreads();

    int wave = tid >> 5;
    int lane = tid & 31;
    int half = lane >> 4;   // selects K-pair {0,1} vs {2,3} in A and B fragments
    int m    = lane & 15;   // point-row within tile (A-matrix M), also D column N

    for (int tile = blockIdx.x * WAVES_PER_BLOCK + wave; tile < ntiles;
         tile += gridDim.x * WAVES_PER_BLOCK) {
        int p  = tile * 16 + m;
        int pc = p < npts ? p : npts - 1;         // clamp (keeps EXEC uniform)
        float x0 = x[2 * pc + 0];
        float x1 = x[2 * pc + 1];

        v8f acc = {};                              // C/D accumulator (16x16 f32)

        #pragma unroll 4
        for (int ck = 0; ck < NWAVES / 4; ++ck) {
            int pp = ck * 2 + half;                // this lane-half's wave pair
            float4 q = ab[pp];                     // {A0,B0,A1,B1}
            float4 c = cc[pp];                     // {c0_0,c0_1,c1_0,c1_1}

            float t0 = __builtin_fmaf(x0, q.x, x1 * q.y);
            float r0 = t0 - __builtin_rintf(t0);   // [-0.5, 0.5] revolutions
            float t1 = __builtin_fmaf(x0, q.z, x1 * q.w);
            float r1 = t1 - __builtin_rintf(t1);

            v2f As = { __builtin_amdgcn_sinf(r0), __builtin_amdgcn_sinf(r1) };
            v2f Ac = { __builtin_amdgcn_cosf(r0), __builtin_amdgcn_cosf(r1) };
            v2f Bs = { c.x, c.y };                 // contiguous from the b128 load
            v2f Bc = { c.z, c.w };

            // D += Sin(16x4) * C0(4x16) ; D += Cos(16x4) * C1(4x16)
            acc = __builtin_amdgcn_wmma_f32_16x16x4_f32(
                false, As, false, Bs, (short)0, acc, false, false);
            acc = __builtin_amdgcn_wmma_f32_16x16x4_f32(
                false, Ac, false, Bc, (short)0, acc, false, false);
        }

        // D layout: lane holds rows M = v + 8*half (v = VGPR index), col N = m.
        // All 16 columns are identical, so column-0 lanes write the results.
        if (m == 0) {
            int base = tile * 16 + half * 8;
            #pragma unroll
            for (int v = 0; v < 8; ++v) {
                int idx = base + v;
                if (idx < npts) out[idx] = acc[v];
            }
        }
    }
}

extern "C" void kernel_launch(void* const* d_in, const int* in_sizes, int n_in,
                              void* d_out, int out_size, void* d_ws, size_t ws_size,
                              hipStream_t stream) {
    const float* x      = (const float*)d_in[0];
    const float* freqs  = (const float*)d_in[1];
    const float* rots   = (const float*)d_in[2];
    const float* coeffs = (const float*)d_in[3];
    float* out = (float*)d_out;

    int npts   = in_sizes[0] / 2;                 // 500000
    int ntiles = (npts + 15) / 16;                // 31250 (exact)
    int blocks = (ntiles + WAVES_PER_BLOCK - 1) / WAVES_PER_BLOCK;
    if (blocks > 16384) blocks = 16384;

    periodic2d_wmma_kernel<<<blocks, BLOCK, 0, stream>>>(
        x, freqs, rots, coeffs, out, npts, ntiles);
}